// VolumeNormalizer_54503134986656
// MI455X (gfx1250) — compile-verified
//
#include <hip/hip_runtime.h>
#include <hip/hip_bf16.h>

typedef __attribute__((ext_vector_type(2))) float v2f;
typedef __attribute__((ext_vector_type(4))) float v4f;
typedef __attribute__((ext_vector_type(8))) float v8f;

#define NVERTS 100000
#define NTRIS  200000
#define NBATCH 64
#define VPB    300000   // floats per batch = 3*NVERTS
#define NBLK_A 256      // reduction blocks per batch
#define TPB    256      // threads per block (8 waves of 32)

// ---- wave32 sum using V_WMMA_F32_16X16X4_F32 as a reduction primitive ----
// A[m][k]: lane m holds K=0 in VGPR0, lane m+16 holds K=2 in VGPR0 (K=1,3 = 0).
// With B = ones, D[m][n] = s_m + s_{m+16} for every n. Summing the 8 D VGPRs
// gives lanes 0-15: sum(t_0..t_7), lanes 16-31: sum(t_8..t_15); one xor-16
// shuffle completes the full 32-lane sum in all lanes.
__device__ __forceinline__ float wave32_sum_wmma(float s) {
  v2f a;     a[0] = s;   a[1] = 0.0f;
  v2f ones;  ones[0] = 1.0f; ones[1] = 1.0f;
  v8f c = {};
  v8f d = __builtin_amdgcn_wmma_f32_16x16x4_f32(
      /*neg_a=*/false, a, /*neg_b=*/false, ones,
      /*c_mod=*/(short)0, c, /*reuse_a=*/false, /*reuse_b=*/false);
  float u = d[0] + d[1] + d[2] + d[3] + d[4] + d[5] + d[6] + d[7];
  u += __shfl_xor(u, 16, 32);
  return u;
}

// ---- Kernel A: per-block partial sums of |det3x3| over gathered triples ----
__global__ void vol_partial_kernel(const float* __restrict__ x,
                                   const long long* __restrict__ M,
                                   float* __restrict__ partials) {
  const int b   = blockIdx.y;
  const int tid = threadIdx.x;
  const float* __restrict__ xb = x + (size_t)b * VPB;

  float s = 0.0f;
  for (int i = blockIdx.x * TPB + tid; i < NTRIS; i += NBLK_A * TPB) {
    const long long base = 3LL * (long long)i;
    const int i0 = (int)M[base + 0];
    const int i1 = (int)M[base + 1];
    const int i2 = (int)M[base + 2];
    const float* p0 = xb + 3 * i0;   // row 0 of the 3x3
    const float* p1 = xb + 3 * i1;   // row 1
    const float* p2 = xb + 3 * i2;   // row 2
    const float a00 = p0[0], a01 = p0[1], a02 = p0[2];
    const float a10 = p1[0], a11 = p1[1], a12 = p1[2];
    const float a20 = p2[0], a21 = p2[1], a22 = p2[2];
    const float det = a00 * (a11 * a22 - a12 * a21)
                    - a01 * (a10 * a22 - a12 * a20)
                    + a02 * (a10 * a21 - a11 * a20);
    s += fabsf(det);
  }

  // All 256 threads reconverge here -> EXEC all ones for WMMA.
  const float w = wave32_sum_wmma(s);

  __shared__ float lsum[TPB / 32];
  if ((tid & 31) == 0) lsum[tid >> 5] = w;
  __syncthreads();
  if (tid == 0) {
    float p = 0.0f;
#pragma unroll
    for (int k = 0; k < TPB / 32; ++k) p += lsum[k];
    partials[(size_t)b * NBLK_A + blockIdx.x] = p;
  }
}

// ---- Kernel B: deterministic tree reduction of partials -> 1/cbrt(vol) ----
__global__ void scale_kernel(const float* __restrict__ partials,
                             float* __restrict__ inv_scale) {
  const int b   = blockIdx.x;
  const int tid = threadIdx.x;      // NBLK_A threads
  __shared__ float sm[NBLK_A];
  sm[tid] = partials[(size_t)b * NBLK_A + tid];
  __syncthreads();
  for (int off = NBLK_A / 2; off > 0; off >>= 1) {
    if (tid < off) sm[tid] += sm[tid + off];
    __syncthreads();
  }
  if (tid == 0) {
    const float vol = sm[0] * (1.0f / 6.0f);
    inv_scale[b] = 1.0f / cbrtf(vol);
  }
}

// ---- Kernel C: out = x * inv_scale[b], float4 loads + NT stores ----------
__global__ void scale_apply_kernel(const v4f* __restrict__ x4,
                                   const float* __restrict__ inv_scale,
                                   v4f* __restrict__ out4) {
  const long long i = (long long)blockIdx.x * blockDim.x + threadIdx.x;
  const long long total = (long long)NBATCH * (VPB / 4);
  if (i >= total) return;
  const int b = (int)(i / (VPB / 4));   // 300000 % 4 == 0: float4 never spans batches
  const float inv = inv_scale[b];
  v4f v = x4[i];
  v[0] *= inv; v[1] *= inv; v[2] *= inv; v[3] *= inv;
  // Output is never re-read: non-temporal store keeps L2 free for the
  // gather-resident vertex slabs.
  __builtin_nontemporal_store(v, &out4[i]);
}

extern "C" void kernel_launch(void* const* d_in, const int* in_sizes, int n_in,
                              void* d_out, int out_size, void* d_ws, size_t ws_size,
                              hipStream_t stream) {
  const float*     x = (const float*)d_in[0];        // (64, 300000) f32
  const long long* M = (const long long*)d_in[1];    // (200000, 3) i64
  float* out = (float*)d_out;                        // (64, 300000) f32

  float* partials  = (float*)d_ws;                   // 64 * NBLK_A floats
  float* inv_scale = partials + NBATCH * NBLK_A;     // 64 floats

  dim3 gridA(NBLK_A, NBATCH);
  vol_partial_kernel<<<gridA, TPB, 0, stream>>>(x, M, partials);

  scale_kernel<<<NBATCH, NBLK_A, 0, stream>>>(partials, inv_scale);

  const long long total4 = (long long)NBATCH * (VPB / 4);
  const int blocksC = (int)((total4 + TPB - 1) / TPB);
  scale_apply_kernel<<<blocksC, TPB, 0, stream>>>((const v4f*)x, inv_scale,
                                                  (v4f*)out);
}